// QuanvLayer_23630910063048
// MI455X (gfx1250) — compile-verified
//
#include <hip/hip_runtime.h>

// Quanvolutional layer, analytically collapsed:
//   z_i  = cos(w_i) * cos(x_i)   per patch element (patch = [p00,p01,p10,p11])
//   out0 = z1*z2*z3, out1 = z0*z1, out2 = z0*z1*z2, out3 = z0*z1*z2*z3
//
// Two stream-ordered kernels:
//  1) quanv_prep : 4 threads, accurate cosf(w_k) -> d_ws (runs once, tiny)
//  2) quanv_fused: pure streaming kernel, 50.3 MB total traffic -> ~2.2 us
//     HBM floor at 23.3 TB/s. One thread = 4 horizontally-adjacent patches:
//       loads : 1 x b128 of precomputed weight factors (broadcast)
//               2 x b128 input row 2i + 2 x b128 row 2i+1 (NT hint)
//       stores: 4 x b128, one per output plane (NT hint)
//     Per-lane cosines via hardware v_cos_f32 (__cosf): inputs are N(0,1),
//     well inside its accurate range; 2 VALU per cosine keeps compute fully
//     hidden under the memory stream.

typedef __attribute__((ext_vector_type(4))) float f4;

#define IN_W   128
#define IN_HW  (128 * 128)   // per (b,c) input plane
#define OUT_W  64
#define OUT_HW (64 * 64)     // per output plane (4096 floats)

__global__ __launch_bounds__(32) void quanv_prep(const float* __restrict__ wts,
                                                 float* __restrict__ cw) {
    int t = threadIdx.x;
    if (t < 4) cw[t] = cosf(wts[t]);   // accurate OCML cos, 4 lanes, runs once
}

__global__ __launch_bounds__(256) void quanv_fused(const float* __restrict__ x,
                                                   const float* __restrict__ cwp,
                                                   float* __restrict__ out,
                                                   int nthreads) {
    int tid = blockIdx.x * 256 + threadIdx.x;
    if (tid >= nthreads) return;

    // Precomputed per-qubit weight factors cos(w_k): one broadcast b128 load.
    const f4 cw = *(const f4*)cwp;

    const int j4 = tid & 15;          // group of 4 patches: input cols 8*j4 .. 8*j4+7
    const int i  = (tid >> 4) & 63;   // patch row (input rows 2i, 2i+1)
    const int bc = tid >> 10;         // fused batch*channel index

    const float* r0 = x + (size_t)bc * IN_HW + (size_t)(2 * i) * IN_W + 8 * j4;
    const f4* p0 = (const f4*)r0;            // 32B aligned
    const f4* p1 = (const f4*)(r0 + IN_W);   // next row, 512B apart

    f4 a0 = __builtin_nontemporal_load(p0);
    f4 a1 = __builtin_nontemporal_load(p0 + 1);
    f4 b0 = __builtin_nontemporal_load(p1);
    f4 b1 = __builtin_nontemporal_load(p1 + 1);

    // row-major patch order: x0=top[2p], x1=top[2p+1], x2=bot[2p], x3=bot[2p+1]
    float top[8] = {a0[0], a0[1], a0[2], a0[3], a1[0], a1[1], a1[2], a1[3]};
    float bot[8] = {b0[0], b0[1], b0[2], b0[3], b1[0], b1[1], b1[2], b1[3]};

    f4 o0, o1, o2, o3;
#pragma unroll
    for (int p = 0; p < 4; ++p) {
        // __cosf -> v_mul_f32 (1/2pi) + v_cos_f32 : hardware transcendental
        float z0 = cw[0] * __cosf(top[2 * p]);
        float z1 = cw[1] * __cosf(top[2 * p + 1]);
        float z2 = cw[2] * __cosf(bot[2 * p]);
        float z3 = cw[3] * __cosf(bot[2 * p + 1]);
        float z01  = z0 * z1;
        float z012 = z01 * z2;
        o0[p] = z1 * (z2 * z3);   // <Z0> = z1*z2*z3
        o1[p] = z01;              // <Z1> = z0*z1
        o2[p] = z012;             // <Z2> = z0*z1*z2
        o3[p] = z012 * z3;        // <Z3> = z0*z1*z2*z3
    }

    // Output planes: out[((bc)*4 + k)*4096 + i*64 + 4*j4], 16B aligned.
    float* ob = out + (size_t)bc * (4 * OUT_HW) + (size_t)i * OUT_W + 4 * j4;
    __builtin_nontemporal_store(o0, (f4*)(ob));
    __builtin_nontemporal_store(o1, (f4*)(ob + OUT_HW));
    __builtin_nontemporal_store(o2, (f4*)(ob + 2 * OUT_HW));
    __builtin_nontemporal_store(o3, (f4*)(ob + 3 * OUT_HW));
}

extern "C" void kernel_launch(void* const* d_in, const int* in_sizes, int n_in,
                              void* d_out, int out_size, void* d_ws, size_t ws_size,
                              hipStream_t stream) {
    const float* x   = (const float*)d_in[0];   // (B,3,128,128) f32
    const float* wts = (const float*)d_in[1];   // (1,4) f32
    float* out = (float*)d_out;                 // (B,12,64,64) f32
    float* cw  = (float*)d_ws;                  // 4 precomputed cos(w_k)

    const int bcCount  = in_sizes[0] / IN_HW;          // B*C = 384
    const int nthreads = bcCount * 64 * 16;            // 1 thread per 4 patches
    const int blocks   = (nthreads + 255) / 256;       // 1536 blocks of 8 wave32

    quanv_prep<<<1, 32, 0, stream>>>(wts, cw);
    quanv_fused<<<blocks, 256, 0, stream>>>(x, cw, out, nthreads);
}